// AbstractFullyConnected_5437428597467
// MI455X (gfx1250) — compile-verified
//
#include <hip/hip_runtime.h>

typedef float v2f __attribute__((ext_vector_type(2)));
typedef float v4f __attribute__((ext_vector_type(4)));
typedef float v8f __attribute__((ext_vector_type(8)));

#define MEAN_C  0.1307f
#define SIGMA_C 0.3081f
#define EPS_C   1e-7f

#define VSTRIDE 2064          // per-vector LDS stride (floats); >= max Kper (2048)
#define CHUNK   256           // columns per async-staged W tile
#define TSTRIDE 260           // padded tile row stride (floats): bank-conflict rotation
#define TFLOATS (16 * TSTRIDE)

__global__ void normalize_kernel(const float* __restrict__ x,
                                 const float* __restrict__ low,
                                 const float* __restrict__ high,
                                 float* __restrict__ ox,
                                 float* __restrict__ olo,
                                 float* __restrict__ ohi,
                                 int n) {
  int i = blockIdx.x * blockDim.x + threadIdx.x;
  if (i < n) {
    ox[i]  = (x[i]    - MEAN_C) / SIGMA_C;
    olo[i] = (low[i]  - MEAN_C) / SIGMA_C;
    ohi[i] = (high[i] - MEAN_C) / SIGMA_C;
  }
}

// Block (rb, slice) computes 16-row K-slice partials of
//   sx = W@x, sm = W@mid, sa = |W|@rad
// W tiles are DMA'd to LDS with global_load_async_to_lds_b128 (double-buffered,
// ASYNCcnt-tracked); compute is pure LDS->WMMA (f32 16x16x4).
__launch_bounds__(256)
__global__ void abox_partial_kernel(const float* __restrict__ W,
                                    const float* __restrict__ in_x,
                                    const float* __restrict__ in_low,
                                    const float* __restrict__ in_high,
                                    float* __restrict__ partial,
                                    int M, int K, int kslices, int Kper) {
  __shared__ float sVec[3 * VSTRIDE];    // [x | mid | rad], indexed by (k - k0)
  __shared__ float tileA[2 * TFLOATS];   // double-buffered 16 x CHUNK W tile
  __shared__ float red[8][3][16];        // [wave][{x,mid,rad}][row]

  const int tid   = threadIdx.x;
  const int rb    = blockIdx.x / kslices;
  const int slice = blockIdx.x % kslices;
  const int k0    = slice * Kper;
  const int k1    = (k0 + Kper < K) ? (k0 + Kper) : K;
  const int Keff  = k1 - k0;

  // Stage this slice's vectors into LDS (x, mid, rad derived here).
  for (int i = k0 + tid; i < k1; i += 256) {
    float lo = in_low[i], hi = in_high[i];
    sVec[i - k0]               = in_x[i];
    sVec[VSTRIDE + (i - k0)]     = 0.5f * (lo + hi);
    sVec[2 * VSTRIDE + (i - k0)] = 0.5f * (hi - lo);
  }

  // ---- async W-tile staging setup ----
  const int ar  = tid & 15;    // row this thread DMAs
  const int ac4 = tid >> 4;    // base b128 index within the row (0..15), +16 per j
  int mg_a = rb * 16 + ar;
  if (mg_a >= M) mg_a = 0;     // clamp: garbage rows land in unread D rows
  const float* __restrict__ arow = W + (size_t)mg_a * (size_t)K + k0;
  const unsigned lds_tile0 = (unsigned)(size_t)(&tileA[0]);

  // Issue one 16 x cols chunk into buffer `buf` (cols multiple of 4).
  auto issue_chunk = [&](int buf, int cb, int cols) {
    const unsigned lbase = lds_tile0 + (unsigned)(buf * TFLOATS + ar * TSTRIDE) * 4u;
    #pragma unroll
    for (int j = 0; j < 4; ++j) {
      const int c4 = ac4 + 16 * j;        // b128 index within row
      if (c4 * 4 < cols) {
        const float* gp = arow + cb + c4 * 4;
        const unsigned lp = lbase + (unsigned)(c4 * 16);
        asm volatile("global_load_async_to_lds_b128 %0, %1, off"
                     :: "v"(lp), "v"(gp) : "memory");
      }
    }
  };

  // ---- WMMA lane mapping ----
  const int wave = tid >> 5;
  const int lane = tid & 31;
  const int ncol = lane & 15;   // D/B column this lane feeds
  const int half = lane >> 4;   // K sub-half
  const int mrow = lane & 15;
  const int b1off = (ncol == 1) ? VSTRIDE : 0;   // col1 -> mid, else x (cols>=2: garbage, unread)
  const int b2off = 2 * VSTRIDE;                 // rad

  v8f acc1 = {};   // col0: W@x   col1: W@mid
  v8f acc2 = {};   // col0: |W|@rad

  // Prologue: first chunk in flight, then vectors+tile visible after barrier.
  const int cols0 = (Keff < CHUNK) ? Keff : CHUNK;
  issue_chunk(0, 0, cols0);
  asm volatile("s_wait_asynccnt 0x0" ::: "memory");
  __syncthreads();

  int buf = 0;
  for (int cb = 0; cb < Keff; cb += CHUNK) {
    const int cols = (Keff - cb < CHUNK) ? (Keff - cb) : CHUNK;
    const int ncb  = cb + CHUNK;
    if (ncb < Keff) {
      const int ncols = (Keff - ncb < CHUNK) ? (Keff - ncb) : CHUNK;
      issue_chunk(buf ^ 1, ncb, ncols);
    }

    const float* __restrict__ tp0 = tileA + buf * TFLOATS + mrow * TSTRIDE + 4 * half;
    // Each wave covers column groups lc = wave*8, wave*8+64, ... within the chunk.
    for (int lc = wave * 8; lc + 8 <= cols; lc += 64) {
      v4f a  = *(const v4f*)(tp0 + lc);
      const int bi = cb + lc + 4 * half;
      v4f b1 = *(const v4f*)(sVec + b1off + bi);
      v4f b2 = *(const v4f*)(sVec + b2off + bi);

      v2f A1  = {a.x, a.y};
      v2f A2  = {a.z, a.w};
      v2f AA1 = {__builtin_fabsf(a.x), __builtin_fabsf(a.y)};
      v2f AA2 = {__builtin_fabsf(a.z), __builtin_fabsf(a.w)};
      v2f B1a = {b1.x, b1.y};
      v2f B1b = {b1.z, b1.w};
      v2f B2a = {b2.x, b2.y};
      v2f B2b = {b2.z, b2.w};

      acc1 = __builtin_amdgcn_wmma_f32_16x16x4_f32(false, A1,  false, B1a,
                                                   (short)0, acc1, false, false);
      acc1 = __builtin_amdgcn_wmma_f32_16x16x4_f32(false, A2,  false, B1b,
                                                   (short)0, acc1, false, false);
      acc2 = __builtin_amdgcn_wmma_f32_16x16x4_f32(false, AA1, false, B2a,
                                                   (short)0, acc2, false, false);
      acc2 = __builtin_amdgcn_wmma_f32_16x16x4_f32(false, AA2, false, B2b,
                                                   (short)0, acc2, false, false);
    }

    // Drain own async transfers, then make everyone's visible & release buffers.
    asm volatile("s_wait_asynccnt 0x0" ::: "memory");
    __syncthreads();
    buf ^= 1;
  }

  // D layout: VGPR r, lanes 0-15 -> (M=r, N=lane); lanes 16-31 -> (M=8+r, N=lane-16).
  if (lane == 0) {
    #pragma unroll
    for (int r = 0; r < 8; ++r) { red[wave][0][r] = acc1[r]; red[wave][2][r] = acc2[r]; }
  } else if (lane == 16) {
    #pragma unroll
    for (int r = 0; r < 8; ++r) { red[wave][0][8 + r] = acc1[r]; red[wave][2][8 + r] = acc2[r]; }
  } else if (lane == 1) {
    #pragma unroll
    for (int r = 0; r < 8; ++r) { red[wave][1][r] = acc1[r]; }
  } else if (lane == 17) {
    #pragma unroll
    for (int r = 0; r < 8; ++r) { red[wave][1][8 + r] = acc1[r]; }
  }
  __syncthreads();

  if (tid < 16) {
    const int row = rb * 16 + tid;
    if (row < M) {
      float sx = 0.f, sm = 0.f, sa = 0.f;
      #pragma unroll
      for (int w = 0; w < 8; ++w) {
        sx += red[w][0][tid];
        sm += red[w][1][tid];
        sa += red[w][2][tid];
      }
      float* pp = partial + (size_t)slice * 3 * (size_t)M;
      pp[row]         = sx;
      pp[M + row]     = sm;
      pp[2 * M + row] = sa;
    }
  }
}

// Sum K-slice partials, add bias, apply DeepPoly ReLU box, emit x/low/high.
__global__ void combine_kernel(const float* __restrict__ partial,
                               const float* __restrict__ bias,
                               float* __restrict__ out_x,
                               float* __restrict__ out_low,
                               float* __restrict__ out_high,
                               int M, int kslices, int apply_relu) {
  const int row = blockIdx.x * blockDim.x + threadIdx.x;
  if (row >= M) return;
  float sx = 0.f, sm = 0.f, sa = 0.f;
  for (int s = 0; s < kslices; ++s) {
    const float* pp = partial + (size_t)s * 3 * (size_t)M;
    sx += pp[row];
    sm += pp[M + row];
    sa += pp[2 * M + row];
  }
  const float bb = bias[row];
  float xo = sx + bb;
  float lo = sm - sa + bb;
  float hi = sm + sa + bb;
  if (apply_relu) {
    xo = fmaxf(xo, 0.f);
    if (hi <= 0.f) {                  // dead
      lo = 0.f; hi = 0.f;
    } else if (lo < 0.f) {            // crossing: DeepPoly upper, lambda=0 lower
      const float denom = hi - lo;
      const float slope = hi / (denom + EPS_C);
      const float ubint = -(lo * hi) / denom;
      hi = slope * hi + ubint;
      lo = 0.f;
    }                                 // else: stable positive, unchanged
  }
  out_x[row]    = xo;
  out_low[row]  = lo;
  out_high[row] = hi;
}

static inline void run_layer(const float* W, const float* b,
                             const float* ix, const float* il, const float* ih,
                             float* ox, float* ol, float* oh,
                             float* partial, int M, int K, int kslices,
                             int apply_relu, hipStream_t stream) {
  const int Kper = K / kslices;        // caller guarantees divisible & %8==0
  const int rowblocks = (M + 15) / 16;
  abox_partial_kernel<<<rowblocks * kslices, 256, 0, stream>>>(
      W, ix, il, ih, partial, M, K, kslices, Kper);
  combine_kernel<<<(M + 255) / 256, 256, 0, stream>>>(
      partial, b, ox, ol, oh, M, kslices, apply_relu);
}

extern "C" void kernel_launch(void* const* d_in, const int* in_sizes, int n_in,
                              void* d_out, int out_size, void* d_ws, size_t ws_size,
                              hipStream_t stream) {
  const float* x    = (const float*)d_in[0];
  const float* low  = (const float*)d_in[1];
  const float* high = (const float*)d_in[2];
  const float* W1 = (const float*)d_in[3];  const float* b1 = (const float*)d_in[4];
  const float* W2 = (const float*)d_in[5];  const float* b2 = (const float*)d_in[6];
  const float* W3 = (const float*)d_in[7];  const float* b3 = (const float*)d_in[8];
  const float* W4 = (const float*)d_in[9];  const float* b4 = (const float*)d_in[10];

  const int n0 = in_sizes[0];                 // 784
  const int M1 = in_sizes[4];                 // 4096
  const int K1 = in_sizes[3] / M1;            // 784
  const int M2 = in_sizes[6];                 // 4096
  const int K2 = in_sizes[5] / M2;            // 4096
  const int M3 = in_sizes[8];                 // 4096
  const int K3 = in_sizes[7] / M3;            // 4096
  const int M4 = in_sizes[10];                // 10
  const int K4 = in_sizes[9] / M4;            // 4096

  float* ws = (float*)d_ws;
  float* Ax = ws;             float* Al = ws + 4096;   float* Ah = ws + 8192;
  float* Bx = ws + 12288;     float* Bl = ws + 16384;  float* Bh = ws + 20480;
  float* partial = ws + 24576;   // up to kslices*3*M floats (<= 24576)

  normalize_kernel<<<(n0 + 255) / 256, 256, 0, stream>>>(x, low, high, Ax, Al, Ah, n0);

  // K=784 -> 2 slices of 392 (392 % 8 == 0); K=4096 -> 2 slices of 2048.
  run_layer(W1, b1, Ax, Al, Ah, Bx, Bl, Bh, partial, M1, K1, 2, 1, stream);
  run_layer(W2, b2, Bx, Bl, Bh, Ax, Al, Ah, partial, M2, K2, 2, 1, stream);
  run_layer(W3, b3, Ax, Al, Ah, Bx, Bl, Bh, partial, M3, K3, 2, 1, stream);

  float* out = (float*)d_out;
  // M=10: a single 16-row block; split K=4096 into 8 slices of 512 for parallelism.
  run_layer(W4, b4, Bx, Bl, Bh, out, out + M4, out + 2 * M4, partial,
            M4, K4, 8, 0, stream);
}